// Speedometer_74466142978198
// MI455X (gfx1250) — compile-verified
//
#include <hip/hip_runtime.h>

typedef __attribute__((ext_vector_type(2))) float v2f;
typedef __attribute__((ext_vector_type(8))) float v8f;

constexpr int   B_ = 32;
constexpr int   T_ = 1024;   // sequence length (K of final GEMM)
constexpr int   D_ = 512;    // feature dim    (N of final GEMM)
constexpr int   S_ = 1024;   // num_steps      (M of final GEMM)
constexpr float EPS_ALPHA = 1e-7f;
constexpr float EPS_STEP  = 1e-6f;
constexpr float LOG2E     = 1.4426950408889634f;

// ---------------------------------------------------------------------------
// Kernel 1: sp[b,t] = sigmoid(dot(x[b,t,:], W) + bias). One wave per row.
// ---------------------------------------------------------------------------
__global__ void __launch_bounds__(256)
speeds_kernel(const float* __restrict__ x, const float* __restrict__ W,
              const float* __restrict__ bias, float* __restrict__ sp) {
  const int lane = threadIdx.x & 31;
  const int row  = (blockIdx.x * blockDim.x + threadIdx.x) >> 5;  // b*T+t
  const float4* xr = reinterpret_cast<const float4*>(x + (size_t)row * D_);
  const float4* wr = reinterpret_cast<const float4*>(W);
  float acc = 0.0f;
#pragma unroll
  for (int j = 0; j < D_ / 128; ++j) {           // 512/4 float4 / 32 lanes
    float4 xv = xr[j * 32 + lane];
    float4 wv = wr[j * 32 + lane];
    acc += xv.x * wv.x + xv.y * wv.y + xv.z * wv.z + xv.w * wv.w;
  }
#pragma unroll
  for (int off = 16; off; off >>= 1) acc += __shfl_xor(acc, off, 32);
  if (lane == 0) {
    float z = acc + bias[0];
    float e = __builtin_amdgcn_exp2f(-z * LOG2E);   // v_exp_f32
    sp[row] = 1.0f / (1.0f + e);
  }
}

// ---------------------------------------------------------------------------
// Kernel 2: the sequential scan. One workgroup (1024 thr = 32 waves) / batch.
// ---------------------------------------------------------------------------
__device__ __forceinline__ float block_sum(float v, float* sh_red,
                                           float* sh_bcast, int lane, int wave) {
#pragma unroll
  for (int off = 16; off; off >>= 1) v += __shfl_xor(v, off, 32);
  if (lane == 0) sh_red[wave] = v;
  __syncthreads();
  if (wave == 0) {
    float w = sh_red[lane];
#pragma unroll
    for (int off = 16; off; off >>= 1) w += __shfl_xor(w, off, 32);
    if (lane == 0) *sh_bcast = w;
  }
  __syncthreads();
  return *sh_bcast;
}

__global__ void __launch_bounds__(1024)
scan_kernel(const float* __restrict__ sp_all, float* __restrict__ alphas) {
  __shared__ float sh_alpha[T_];
  __shared__ float sh_red[32];
  __shared__ float sh_bcast;
  const int b    = blockIdx.x;
  const int t    = threadIdx.x;
  const int lane = t & 31;
  const int wave = t >> 5;

  const float sp = sp_all[b * T_ + t];
  float a = (t == 0) ? 1.0f : EPS_ALPHA;
  sh_alpha[t] = a;
  __syncthreads();

  float* out = alphas + (size_t)b * S_ * T_ + t;   // layout (B, S, T)
  for (int step = 0; step < S_; ++step) {
    const float u    = block_sum(a * sp, sh_red, &sh_bcast, lane, wave);
    const float sq   = 1.7f - fabsf(u - 0.5f);
    const float prev = (t == 0) ? 0.0f : sh_alpha[t - 1];
    const float base = (1.0f - u) * a + u * prev + EPS_STEP;
    // base > 0 always: pow(base, sq) = exp2(sq * log2(base))
    const float an = __builtin_amdgcn_exp2f(sq * __builtin_amdgcn_logf(base));
    const float s  = block_sum(an, sh_red, &sh_bcast, lane, wave);
    a = an / s;
    out[(size_t)step * T_] = a;
    sh_alpha[t] = a;   // barrier-separated from all reads (inside block_sum)
  }
}

// ---------------------------------------------------------------------------
// Kernel 3: contexts[b] = alphas[b](SxT) @ x[b](TxD) via V_WMMA_F32_16X16X4_F32
// Block (8 waves) -> 128(M) x 64(N) tile; x chunk (64K x 64N) staged in LDS
// (row stride 72 floats: conflict-free frag reads, 16B-aligned b128 stores),
// register double-buffered so next chunk's global loads overlap the WMMAs.
// A frag (ISA 7.12.2, 32-bit A 16x4): lanes 0-15 K={k,k+1} of row M=lane;
// lanes 16-31 K={k+2,k+3}. C/D: VGPR v -> M=v (lanes 0-15), M=v+8 (16-31).
// ---------------------------------------------------------------------------
__global__ void __launch_bounds__(256)
ctx_gemm_kernel(const float* __restrict__ alphas, const float* __restrict__ x,
                float* __restrict__ ctx) {
  __shared__ float Bs[64][72];            // 18 KB, padded stride
  const int tid  = threadIdx.x;
  const int lane = tid & 31;
  const int wav  = tid >> 5;              // 0..7
  const int hl   = lane >> 4;             // lane half
  const int l16  = lane & 15;

  // blockIdx.x = ((b*8 + mblk)*8 + ng);  32*8*8 = 2048 blocks
  const int ng   = blockIdx.x & 7;        // 64-col N group
  const int mblk = (blockIdx.x >> 3) & 7; // 128-row M block
  const int b    = blockIdx.x >> 6;

  const int n0 = ng * 64;
  const float* Arow  = alphas + ((size_t)b * S_ + mblk * 128 + wav * 16 + l16) * T_;
  const float* Xb    = x + (size_t)b * T_ * D_ + n0;
  float*       Cbase = ctx + ((size_t)b * S_ + mblk * 128 + wav * 16) * D_ + n0;

  // cooperative copy indices: 256 thr move a 64x64 chunk (16 floats each)
  const int cr = tid >> 2;                // 0..63 : chunk row
  const int cf = (tid & 3) * 4;           // col base within 16-wide group

  float4 stage[4];
#pragma unroll
  for (int j = 0; j < 4; ++j)             // preload chunk 0
    stage[j] = *reinterpret_cast<const float4*>(Xb + (size_t)cr * D_ + 16 * j + cf);

  v8f acc[4] = {};
  constexpr int NC = T_ / 64;             // 16 K-chunks
  for (int c = 0; c < NC; ++c) {
#pragma unroll
    for (int j = 0; j < 4; ++j)           // LDS fill (ds_store_b128)
      *reinterpret_cast<float4*>(&Bs[cr][16 * j + cf]) = stage[j];
    __syncthreads();

    if (c + 1 < NC) {                     // issue next chunk's loads early
      const float* Xn = Xb + (size_t)(c + 1) * 64 * D_;
#pragma unroll
      for (int j = 0; j < 4; ++j)
        stage[j] = *reinterpret_cast<const float4*>(Xn + (size_t)cr * D_ + 16 * j + cf);
    }

    const float* Ac = Arow + c * 64;
#pragma unroll 4
    for (int kk = 0; kk < 64; kk += 4) {
      const float2 av = *reinterpret_cast<const float2*>(Ac + kk + 2 * hl);
      v2f af; af.x = av.x; af.y = av.y;
      const int r0 = kk + 2 * hl;
#pragma unroll
      for (int j = 0; j < 4; ++j) {
        v2f bf;
        bf.x = Bs[r0][j * 16 + l16];
        bf.y = Bs[r0 + 1][j * 16 + l16];
        acc[j] = __builtin_amdgcn_wmma_f32_16x16x4_f32(
            false, af, false, bf, (short)0, acc[j], false, false);
      }
    }
    __syncthreads();                      // protect Bs before next fill
  }

#pragma unroll
  for (int j = 0; j < 4; ++j) {
#pragma unroll
    for (int v = 0; v < 8; ++v) {
      const int m = v + 8 * hl;
      Cbase[(size_t)m * D_ + j * 16 + l16] = acc[j][v];
    }
  }
}

// ---------------------------------------------------------------------------
extern "C" void kernel_launch(void* const* d_in, const int* in_sizes, int n_in,
                              void* d_out, int out_size, void* d_ws,
                              size_t ws_size, hipStream_t stream) {
  (void)in_sizes; (void)n_in; (void)out_size; (void)ws_size;
  const float* x    = (const float*)d_in[0];   // (B,T,D) f32
  const float* W    = (const float*)d_in[1];   // (D,1)   f32
  const float* bias = (const float*)d_in[2];   // (1,)    f32
  // d_in[3]=r, d_in[4]=num_steps (compile-time constants here)

  float* out    = (float*)d_out;
  float* ctx    = out;                                   // (B,S,D)
  float* alphas = out + (size_t)B_ * S_ * D_;            // (B,S,T)
  float* sp     = (float*)d_ws;                          // (B,T) scratch

  // 1) speeds: one wave per (b,t) row; 32768 waves -> 4096 blocks of 256.
  speeds_kernel<<<(B_ * T_) / 8, 256, 0, stream>>>(x, W, bias, sp);

  // 2) sequential scan: one 1024-thread workgroup per batch.
  scan_kernel<<<B_, 1024, 0, stream>>>(sp, alphas);

  // 3) WMMA f32 GEMM: 32 batches x 8 M-blocks x 8 N-groups = 2048 blocks.
  ctx_gemm_kernel<<<B_ * 8 * 8, 256, 0, stream>>>(alphas, x, ctx);
}